// KnowledgeAttention_9826885173721
// MI455X (gfx1250) — compile-verified
//
#include <hip/hip_runtime.h>
#include <hip/hip_bf16.h>

// Shapes (fixed by the reference)
#define Bq   16
#define Tq   256
#define Kq   128
#define DE   256
#define DK   256
#define Uq   128

typedef __attribute__((ext_vector_type(2))) float v2f;
typedef __attribute__((ext_vector_type(8))) float v8f;

// ---------------------------------------------------------------------------
// Generic fp32 WMMA GEMM with bias:  C[M,N] = A[M,Kd] @ Bm[Kd,N] + bias[N]
// One wave32 computes one 16x16 tile via V_WMMA_F32_16X16X4_F32, K step 4.
// Layouts (CDNA5 ISA 7.12.2, 32-bit):
//   A 16x4 : lane<16 holds M=lane, {K=kk, K=kk+1}; lane>=16 holds {K=kk+2, kk+3}
//   B 4x16 : vgpr0 = B[kk + 2*(lane>>4)][lane&15], vgpr1 = next K row
//   C 16x16: vgpr r = C[r + 8*(lane>>4)][lane&15]
// Launch with exactly (Mtiles*Ntiles) waves, 8 waves (256 thr) per block.
// ---------------------------------------------------------------------------
__global__ __launch_bounds__(256)
void gemm_bias_wmma_f32(const float* __restrict__ A,
                        const float* __restrict__ Bm,
                        const float* __restrict__ bias,
                        float* __restrict__ C,
                        int M, int Kd, int N) {
    const int lane  = threadIdx.x & 31;
    const int wave  = blockIdx.x * 8 + (threadIdx.x >> 5);
    const int tn_ct = N >> 4;
    const int tm    = wave / tn_ct;
    const int tn    = wave % tn_ct;

    const int row   = tm * 16 + (lane & 15);   // A row for this lane
    const int col   = tn * 16 + (lane & 15);   // B/C column for this lane
    const int khalf = (lane >> 4) << 1;        // 0 or 2

    v8f c = {};
    for (int kk = 0; kk < Kd; kk += 4) {
        const int k0 = kk + khalf;
        v2f a, b;
        a.x = A[row * Kd + k0];
        a.y = A[row * Kd + k0 + 1];
        b.x = Bm[k0 * N + col];
        b.y = Bm[(k0 + 1) * N + col];
        c = __builtin_amdgcn_wmma_f32_16x16x4_f32(false, a, false, b,
                                                  (short)0, c, false, false);
    }

    const float bv    = bias[col];
    const int   mbase = tm * 16 + ((lane >> 4) << 3);
#pragma unroll
    for (int r = 0; r < 8; ++r)
        C[(mbase + r) * N + col] = c[r] + bv;
}

// ---------------------------------------------------------------------------
// Batched fp32 WMMA GEMM: out[b] = attn_b[Tq,Kq] @ knowledge_b[Kq,DK]
// 256 tiles per batch, 16 batches -> 4096 waves -> 512 blocks of 8 waves.
// ---------------------------------------------------------------------------
__global__ __launch_bounds__(256)
void context_wmma_f32(const float* __restrict__ attn,
                      const float* __restrict__ knowledge,
                      float* __restrict__ out) {
    const int lane = threadIdx.x & 31;
    const int wave = blockIdx.x * 8 + (threadIdx.x >> 5);
    const int b    = wave >> 8;           // 256 tiles per batch
    const int tile = wave & 255;
    const int tm   = tile >> 4;           // Tq/16 = 16 row tiles
    const int tn   = tile & 15;           // DK/16 = 16 col tiles

    const float* __restrict__ A  = attn      + (size_t)b * Tq * Kq;
    const float* __restrict__ Bm = knowledge + (size_t)b * Kq * DK;
    float* __restrict__       C  = out       + (size_t)b * Tq * DK;

    const int row   = tm * 16 + (lane & 15);
    const int col   = tn * 16 + (lane & 15);
    const int khalf = (lane >> 4) << 1;

    v8f c = {};
    for (int kk = 0; kk < Kq; kk += 4) {
        const int k0 = kk + khalf;
        v2f a, bb;
        a.x  = A[row * Kq + k0];
        a.y  = A[row * Kq + k0 + 1];
        bb.x = Bm[k0 * DK + col];
        bb.y = Bm[(k0 + 1) * DK + col];
        c = __builtin_amdgcn_wmma_f32_16x16x4_f32(false, a, false, bb,
                                                  (short)0, c, false, false);
    }

    const int mbase = tm * 16 + ((lane >> 4) << 3);
#pragma unroll
    for (int r = 0; r < 8; ++r)
        C[(mbase + r) * DK + col] = c[r];
}

// ---------------------------------------------------------------------------
// Fused score + softmax:
//   score[k] = V . tanh(E[b,t,:] + KU[b,k,:]) + bV ; attn = softmax_k(score)
// One block per (b,t); 128 threads; thread tid owns k = tid.
// E row and V staged in LDS; softmax via LDS tree reductions.
// ---------------------------------------------------------------------------
__global__ __launch_bounds__(128)
void score_softmax(const float* __restrict__ E,     // [B*T, U]
                   const float* __restrict__ KU,    // [B*K, U]
                   const float* __restrict__ Vv,    // [U]
                   const float* __restrict__ bV,    // [1]
                   float* __restrict__ attn) {      // [B*T, K]
    __shared__ float s_e[Uq];
    __shared__ float s_v[Uq];
    __shared__ float s_red[Kq];

    const int bt  = blockIdx.x;        // 0..B*T-1
    const int b   = bt >> 8;           // T = 256
    const int tid = threadIdx.x;       // = k

    s_e[tid] = E[(size_t)bt * Uq + tid];
    s_v[tid] = Vv[tid];
    __syncthreads();

    const float* __restrict__ kurow = KU + ((size_t)b * Kq + tid) * Uq;
    float acc = 0.0f;
#pragma unroll 4
    for (int u = 0; u < Uq; u += 4) {
        float4 kq = *reinterpret_cast<const float4*>(kurow + u);
        acc += tanhf(s_e[u + 0] + kq.x) * s_v[u + 0];
        acc += tanhf(s_e[u + 1] + kq.y) * s_v[u + 1];
        acc += tanhf(s_e[u + 2] + kq.z) * s_v[u + 2];
        acc += tanhf(s_e[u + 3] + kq.w) * s_v[u + 3];
    }
    acc += bV[0];

    // softmax over the 128 threads
    s_red[tid] = acc;
    __syncthreads();
    for (int off = 64; off > 0; off >>= 1) {
        if (tid < off) s_red[tid] = fmaxf(s_red[tid], s_red[tid + off]);
        __syncthreads();
    }
    const float mx = s_red[0];
    __syncthreads();

    const float ex = __expf(acc - mx);
    s_red[tid] = ex;
    __syncthreads();
    for (int off = 64; off > 0; off >>= 1) {
        if (tid < off) s_red[tid] += s_red[tid + off];
        __syncthreads();
    }
    attn[(size_t)bt * Kq + tid] = ex / s_red[0];
}

// ---------------------------------------------------------------------------
// Launch
// ---------------------------------------------------------------------------
extern "C" void kernel_launch(void* const* d_in, const int* in_sizes, int n_in,
                              void* d_out, int out_size, void* d_ws, size_t ws_size,
                              hipStream_t stream) {
    const float* knowledge = (const float*)d_in[0]; // [B,K,DK]
    const float* enc       = (const float*)d_in[1]; // [B,T,DE]
    const float* W1        = (const float*)d_in[2]; // [DE,U]
    const float* b1        = (const float*)d_in[3]; // [U]
    const float* W2        = (const float*)d_in[4]; // [DK,U]
    const float* b2        = (const float*)d_in[5]; // [U]
    const float* Vv        = (const float*)d_in[6]; // [U,1]
    const float* bV        = (const float*)d_in[7]; // [1]
    float*       out       = (float*)d_out;         // [B,T,DK]

    // Workspace layout
    float* E    = (float*)d_ws;                        // [B*T, U]  = 2 MB
    float* KU   = E  + (size_t)Bq * Tq * Uq;           // [B*K, U]  = 1 MB
    float* attn = KU + (size_t)Bq * Kq * Uq;           // [B*T, K]  = 2 MB

    // 1) E = enc @ W1 + b1 : M=4096, Kd=256, N=128 -> 2048 waves -> 256 blocks
    gemm_bias_wmma_f32<<<256, 256, 0, stream>>>(enc, W1, b1, E,
                                                Bq * Tq, DE, Uq);

    // 2) KU = knowledge @ W2 + b2 : M=2048, Kd=256, N=128 -> 1024 waves -> 128 blocks
    gemm_bias_wmma_f32<<<128, 256, 0, stream>>>(knowledge, W2, b2, KU,
                                                Bq * Kq, DK, Uq);

    // 3) attn = softmax_k( V . tanh(E + KU) + bV )
    score_softmax<<<Bq * Tq, Kq, 0, stream>>>(E, KU, Vv, bV, attn);

    // 4) out = attn @ knowledge (batched) : 4096 waves -> 512 blocks
    context_wmma_f32<<<512, 256, 0, stream>>>(attn, knowledge, out);
}